// BloomMaskDistillationLoss_7146825580936
// MI455X (gfx1250) — compile-verified
//
#include <hip/hip_runtime.h>
#include <math.h>

typedef __attribute__((ext_vector_type(2))) float v2f;
typedef __attribute__((ext_vector_type(8))) float v8f;

#define B_N   8192
#define D_K   768
#define EPSF  1e-12f
#define TILE  64
#define KC    32
#define LDA   36   // KC + 4 floats pad -> conflict-free b64 operand reads

// ---------------------------------------------------------------------------
// Kernel 1: n_i = max(||F_i * m_i||, eps)
// ---------------------------------------------------------------------------
__global__ __launch_bounds__(256) void rownorm_kernel(
    const float* __restrict__ femb, const float* __restrict__ qmask,
    float* __restrict__ n_out) {
  __shared__ float red[256];
  const int row = blockIdx.x;
  const float* f = femb  + (size_t)row * D_K;
  const float* m = qmask + (size_t)row * D_K;
  float s = 0.f;
  for (int c = threadIdx.x; c < D_K; c += 256) {
    float v = f[c] * m[c];
    s += v * v;
  }
  red[threadIdx.x] = s;
  __syncthreads();
  for (int off = 128; off > 0; off >>= 1) {
    if (threadIdx.x < off) red[threadIdx.x] += red[threadIdx.x + off];
    __syncthreads();
  }
  if (threadIdx.x == 0) n_out[row] = fmaxf(sqrtf(red[0]), EPSF);
}

// ---------------------------------------------------------------------------
// Kernel 2: fused 3-way GEMM tile + epilogue + block reduction.
// One 64x64 output tile per block; 8 wave32s; each wave owns two 16x16
// subtiles and carries 3 f32 WMMA accumulators per subtile.
// ---------------------------------------------------------------------------
__global__ __launch_bounds__(256) void tile_kernel(
    const float* __restrict__ femb, const float* __restrict__ qmask,
    const float* __restrict__ n_i, float* __restrict__ partial) {
  __shared__ float sA[TILE * LDA];  // F_i chunk
  __shared__ float sP[TILE * LDA];  // F_i * m_i^2 chunk
  __shared__ float sM[TILE * LDA];  // m_i^2 chunk
  __shared__ float sB[TILE * LDA];  // F_j chunk
  __shared__ float red[256];

  const int iBase = blockIdx.y * TILE;
  const int jBase = blockIdx.x * TILE;
  const int tid  = threadIdx.x;
  const int lane = tid & 31;
  const int w    = tid >> 5;
  const int srow = (w & 3) * 16;   // subtile row origin within 64x64 tile
  const int scol = (w >> 2) * 32;  // two 16-col subtiles: scol, scol+16

  v8f cF0 = {}, cF1 = {}, cN0 = {}, cN1 = {}, cQ0 = {}, cQ1 = {};

  // staging decomposition: 8 threads per row (float4), 32 rows per pass
  const int lr = tid >> 3;          // 0..31
  const int lc = (tid & 7) * 4;     // 0,4,...,28

  // WMMA operand addressing (ISA 16x4 f32 A-layout == 4x16 B-layout here)
  const int mrow  = srow + (lane & 15);
  const int koff  = (lane >> 4) * 2;
  const int nrow0 = scol + (lane & 15);
  const int nrow1 = nrow0 + 16;

  for (int k0 = 0; k0 < D_K; k0 += KC) {
    __syncthreads();
    for (int rr = lr; rr < TILE; rr += 32) {
      const size_t giOff = (size_t)(iBase + rr) * D_K + k0 + lc;
      const float4 f  = *reinterpret_cast<const float4*>(femb  + giOff);
      const float4 m  = *reinterpret_cast<const float4*>(qmask + giOff);
      const float4 m2 = make_float4(m.x * m.x, m.y * m.y, m.z * m.z, m.w * m.w);
      const float4 p  = make_float4(f.x * m2.x, f.y * m2.y, f.z * m2.z, f.w * m2.w);
      *reinterpret_cast<float4*>(&sA[rr * LDA + lc]) = f;
      *reinterpret_cast<float4*>(&sP[rr * LDA + lc]) = p;
      *reinterpret_cast<float4*>(&sM[rr * LDA + lc]) = m2;
      const size_t gjOff = (size_t)(jBase + rr) * D_K + k0 + lc;
      *reinterpret_cast<float4*>(&sB[rr * LDA + lc]) =
          *reinterpret_cast<const float4*>(femb + gjOff);
      if (k0 + KC < D_K) {  // pull next chunk toward the WGP (global_prefetch_b8)
        __builtin_prefetch(femb  + giOff + KC, 0, 0);
        __builtin_prefetch(qmask + giOff + KC, 0, 0);
        __builtin_prefetch(femb  + gjOff + KC, 0, 0);
      }
    }
    __syncthreads();

#pragma unroll
    for (int k = 0; k < KC; k += 4) {
      const v2f aF = *reinterpret_cast<const v2f*>(&sA[mrow * LDA + k + koff]);
      const v2f aP = *reinterpret_cast<const v2f*>(&sP[mrow * LDA + k + koff]);
      const v2f aM = *reinterpret_cast<const v2f*>(&sM[mrow * LDA + k + koff]);
      const v2f b0 = *reinterpret_cast<const v2f*>(&sB[nrow0 * LDA + k + koff]);
      const v2f b1 = *reinterpret_cast<const v2f*>(&sB[nrow1 * LDA + k + koff]);
      const v2f q0 = b0 * b0;
      const v2f q1 = b1 * b1;
      cF0 = __builtin_amdgcn_wmma_f32_16x16x4_f32(false, aF, false, b0, (short)0, cF0, false, false);
      cN0 = __builtin_amdgcn_wmma_f32_16x16x4_f32(false, aP, false, b0, (short)0, cN0, false, false);
      cQ0 = __builtin_amdgcn_wmma_f32_16x16x4_f32(false, aM, false, q0, (short)0, cQ0, false, false);
      cF1 = __builtin_amdgcn_wmma_f32_16x16x4_f32(false, aF, false, b1, (short)0, cF1, false, false);
      cN1 = __builtin_amdgcn_wmma_f32_16x16x4_f32(false, aP, false, b1, (short)0, cN1, false, false);
      cQ1 = __builtin_amdgcn_wmma_f32_16x16x4_f32(false, aM, false, q1, (short)0, cQ1, false, false);
    }
  }

  // Epilogue: C layout — VGPR r holds row (r + 8*(lane>=16)), col = lane&15.
  float local = 0.f;
  {
    const int giB = iBase + srow + ((lane >> 4) * 8);
    const int gj0 = jBase + scol + (lane & 15);
    const int gj1 = gj0 + 16;
#pragma unroll
    for (int r = 0; r < 8; ++r) {
      const int gi = giB + r;
      const float ni = n_i[gi];
      float d0 = fabsf(cF0[r] - cN0[r] / (ni * fmaxf(sqrtf(fmaxf(cQ0[r], 0.f)), EPSF)));
      float d1 = fabsf(cF1[r] - cN1[r] / (ni * fmaxf(sqrtf(fmaxf(cQ1[r], 0.f)), EPSF)));
      if (gi == gj0) d0 = 0.f;
      if (gi == gj1) d1 = 0.f;
      local += d0 + d1;
    }
  }

  red[tid] = local;
  __syncthreads();
  for (int off = 128; off > 0; off >>= 1) {
    if (tid < off) red[tid] += red[tid + off];
    __syncthreads();
  }
  if (tid == 0) partial[blockIdx.y * gridDim.x + blockIdx.x] = red[0];
}

// ---------------------------------------------------------------------------
// Kernel 3: deterministic final reduction + scaling.
// ---------------------------------------------------------------------------
__global__ __launch_bounds__(256) void reduce_kernel(
    const float* __restrict__ partial, int n, float* __restrict__ out) {
  __shared__ float red[256];
  float s = 0.f;
  for (int i = threadIdx.x; i < n; i += 256) s += partial[i];
  red[threadIdx.x] = s;
  __syncthreads();
  for (int off = 128; off > 0; off >>= 1) {
    if (threadIdx.x < off) red[threadIdx.x] += red[threadIdx.x + off];
    __syncthreads();
  }
  if (threadIdx.x == 0)
    out[0] = red[0] / ((float)B_N * (float)(B_N - 1));
}

// ---------------------------------------------------------------------------
extern "C" void kernel_launch(void* const* d_in, const int* in_sizes, int n_in,
                              void* d_out, int out_size, void* d_ws, size_t ws_size,
                              hipStream_t stream) {
  (void)in_sizes; (void)n_in; (void)out_size; (void)ws_size;
  const float* femb  = (const float*)d_in[0];
  const float* qmask = (const float*)d_in[1];
  float* out = (float*)d_out;

  float* n_i     = (float*)d_ws;          // B_N floats
  float* partial = n_i + B_N;             // (B/TILE)^2 floats

  rownorm_kernel<<<B_N, 256, 0, stream>>>(femb, qmask, n_i);

  dim3 grid(B_N / TILE, B_N / TILE);
  tile_kernel<<<grid, 256, 0, stream>>>(femb, qmask, n_i, partial);

  const int np = (B_N / TILE) * (B_N / TILE);
  reduce_kernel<<<1, 256, 0, stream>>>(partial, np, out);
}